// FlatDilation1D_6270652252753
// MI455X (gfx1250) — compile-verified
//
#include <hip/hip_runtime.h>

#define K_SIZE 201
#define HALF   100
#define PADDED (K_SIZE + 2 * HALF)   // 401
#define NTHREADS 224                 // 7 wave32s, covers 201 outputs

typedef __attribute__((address_space(1))) int global_int;
typedef __attribute__((address_space(3))) int lds_int;

__global__ __launch_bounds__(NTHREADS)
void FlatDilation1D_kernel(const float* __restrict__ x,
                           const float* __restrict__ scale_p,
                           float* __restrict__ out) {
    // Padded input window and the structuring function, both staged in LDS.
    __shared__ float xs[PADDED];
    __shared__ float hh[K_SIZE];

    const int tid = threadIdx.x;
    const float NEG_INF = -__builtin_inff();

    // Fill -inf pads: [0,100) and [301,401). Disjoint from the async-written
    // region [100,301), so no ds-store / async-store race.
    if (tid < HALF) {
        xs[tid]                 = NEG_INF;
        xs[HALF + K_SIZE + tid] = NEG_INF;
    }

    // Stage x -> xs[HALF + tid] via the CDNA5 async global->LDS path
    // (ASYNCcnt-tracked). One b32 per lane, 201 lanes total (804 bytes).
    if (tid < K_SIZE) {
#if __has_builtin(__builtin_amdgcn_global_load_async_to_lds_b32)
        __builtin_amdgcn_global_load_async_to_lds_b32(
            (global_int*)(x + tid),
            (lds_int*)(&xs[HALF + tid]),
            /*offset=*/0, /*cpol=*/0);
#else
        xs[HALF + tid] = x[tid];
#endif
    }

    // h[k] = -((k-100)/scale)^16 via repeated squaring (matches XLA's
    // integer_pow expansion: x2=x*x; x4=x2*x2; x8=x4*x4; x16=x8*x8).
    const float s = scale_p[0];   // uniform -> s_load + broadcast
    if (tid < K_SIZE) {
        float z   = (float)(tid - HALF);
        float t   = z / s;
        float t2  = t * t;
        float t4  = t2 * t2;
        float t8  = t4 * t4;
        float t16 = t8 * t8;
        hh[tid] = -t16;
    }

    // Drain async LDS writes, then make all LDS init visible workgroup-wide.
#if __has_builtin(__builtin_amdgcn_s_wait_asynccnt)
    __builtin_amdgcn_s_wait_asynccnt(0);
#else
    asm volatile("s_wait_asynccnt 0" ::: "memory");
#endif
    __syncthreads();

    // One output per thread. xs[tid+k] is stride-1 across lanes
    // (bank-conflict-free); hh[k] is wave-uniform (LDS broadcast).
    if (tid < K_SIZE) {
        float acc = NEG_INF;
#pragma unroll 4
        for (int k = 0; k < K_SIZE; ++k) {
            acc = fmaxf(acc, xs[tid + k] + hh[k]);
        }
        out[tid] = acc;
    }
}

extern "C" void kernel_launch(void* const* d_in, const int* in_sizes, int n_in,
                              void* d_out, int out_size, void* d_ws, size_t ws_size,
                              hipStream_t stream) {
    const float* x     = (const float*)d_in[0];
    const float* scale = (const float*)d_in[1];
    float*       out   = (float*)d_out;
    (void)in_sizes; (void)n_in; (void)out_size; (void)d_ws; (void)ws_size;

    // Problem is 804 B in / 804 B out: strictly launch-latency bound.
    // Single workgroup (7 wave32s) on one WGP does the whole thing.
    FlatDilation1D_kernel<<<1, NTHREADS, 0, stream>>>(x, scale, out);
}